// DragGNN_52493090292283
// MI455X (gfx1250) — compile-verified
//
#include <hip/hip_runtime.h>

#define N_NODES 50000
#define N_EDGES 400000
#define N_GRAPHS 16

typedef float v2f __attribute__((ext_vector_type(2)));
typedef float v8f __attribute__((ext_vector_type(8)));

#if defined(__HIP_PLATFORM_AMD__)
#define GATOMIC_ADD(p, v) unsafeAtomicAdd((p), (v))
#else
#define GATOMIC_ADD(p, v) atomicAdd((p), (v))
#endif

// ---------------- degree / norm precompute ----------------

__global__ void k_fill1(float* __restrict__ p, int n) {
    int i = blockIdx.x * blockDim.x + threadIdx.x;
    if (i < n) p[i] = 1.0f;  // self-loop contributes 1 to every in-degree
}

__global__ void k_edge_prep(const int* __restrict__ ei, int* __restrict__ row32,
                            int* __restrict__ col32, float* __restrict__ deg, int E) {
    int e = blockIdx.x * blockDim.x + threadIdx.x;
    if (e < E) {
        int r = ei[e];
        int c = ei[E + e];
        row32[e] = r;
        col32[e] = c;
        GATOMIC_ADD(&deg[c], 1.0f);
    }
}

__global__ void k_rsqrt_inplace(float* __restrict__ d, int n) {
    int i = blockIdx.x * blockDim.x + threadIdx.x;
    if (i < n) d[i] = rsqrtf(d[i]);
}

__global__ void k_norm(const float* __restrict__ dinv, const int* __restrict__ row32,
                       const int* __restrict__ col32, float* __restrict__ nrm, int E) {
    int e = blockIdx.x * blockDim.x + threadIdx.x;
    if (e < E) nrm[e] = dinv[row32[e]] * dinv[col32[e]];
}

// ---------------- weight transpose (Wt[n*K+k] = W[k*F+n]) ----------------

__global__ void k_transpose(const float* __restrict__ W, float* __restrict__ Wt,
                            int K, int logF) {
    int t = blockIdx.x * blockDim.x + threadIdx.x;
    int F = 1 << logF;
    if (t < K * F) {
        int k = t >> logF;
        int n = t & (F - 1);
        Wt[(size_t)n * K + k] = W[t];
    }
}

// ---------------- layer-1 GEMM (K=3, store-bandwidth bound) ----------------

__global__ void k_gemm1(const float* __restrict__ x, const float* __restrict__ W1,
                        float* __restrict__ H) {
    int t = blockIdx.x * blockDim.x + threadIdx.x;  // exact N*512 threads
    int i = t >> 9;
    int f = t & 511;
    const float* xr = x + i * 3;
    H[t] = fmaf(xr[0], W1[f], fmaf(xr[1], W1[512 + f], xr[2] * W1[1024 + f]));
}

// ---------------- fp32 WMMA GEMM: C[N,F] = A[N,K] @ Bt[F,K]^T ----------------

#define WMMA4(a, b, c) \
    __builtin_amdgcn_wmma_f32_16x16x4_f32(false, (a), false, (b), (short)0, (c), false, false)

__global__ __launch_bounds__(256) void k_wmma_gemm_f32(
    const float* __restrict__ A,   // [N x K] row-major
    const float* __restrict__ Bt,  // [F x K] row-major (W transposed)
    float* __restrict__ C,         // [N x F]
    int K, int F) {
    const int lane = threadIdx.x & 31;
    const int wave = threadIdx.x >> 5;
    const int m0 = blockIdx.x << 4;                    // 16 rows per block
    const int n0 = ((blockIdx.y << 3) + wave) << 6;    // 64 cols per wave
    const int rc = lane & 15;                          // A: M row / B: N col
    const int kk = (lane >> 4) << 1;                   // K pair: {0,1} or {2,3}

    const float* aPtr = A + (size_t)(m0 + rc) * K + kk;
    const float* bPtr = Bt + (size_t)(n0 + rc) * K + kk;
    const size_t bStride = (size_t)K << 4;             // 16 columns ahead

    v8f acc0 = {}, acc1 = {}, acc2 = {}, acc3 = {};

    // software pipeline: fragments for step k consumed one iteration after load
    v2f a_c  = *(const v2f*)(aPtr);
    v2f b0_c = *(const v2f*)(bPtr);
    v2f b1_c = *(const v2f*)(bPtr + bStride);
    v2f b2_c = *(const v2f*)(bPtr + 2 * bStride);
    v2f b3_c = *(const v2f*)(bPtr + 3 * bStride);

#pragma unroll 2
    for (int k0 = 4; k0 < K; k0 += 4) {
        v2f a_n  = *(const v2f*)(aPtr + k0);
        v2f b0_n = *(const v2f*)(bPtr + k0);
        v2f b1_n = *(const v2f*)(bPtr + bStride + k0);
        v2f b2_n = *(const v2f*)(bPtr + 2 * bStride + k0);
        v2f b3_n = *(const v2f*)(bPtr + 3 * bStride + k0);
        acc0 = WMMA4(a_c, b0_c, acc0);
        acc1 = WMMA4(a_c, b1_c, acc1);
        acc2 = WMMA4(a_c, b2_c, acc2);
        acc3 = WMMA4(a_c, b3_c, acc3);
        a_c = a_n; b0_c = b0_n; b1_c = b1_n; b2_c = b2_n; b3_c = b3_n;
    }
    acc0 = WMMA4(a_c, b0_c, acc0);
    acc1 = WMMA4(a_c, b1_c, acc1);
    acc2 = WMMA4(a_c, b2_c, acc2);
    acc3 = WMMA4(a_c, b3_c, acc3);

    // C/D layout: VGPR v -> M = v (+8 for lanes 16-31), N = lane&15
    const int crow = m0 + ((lane >> 4) << 3);
    float* cPtr = C + (size_t)crow * F + n0 + rc;
#pragma unroll
    for (int v = 0; v < 8; ++v) {
        cPtr[(size_t)v * F + 0]  = acc0[v];
        cPtr[(size_t)v * F + 16] = acc1[v];
        cPtr[(size_t)v * F + 32] = acc2[v];
        cPtr[(size_t)v * F + 48] = acc3[v];
    }
}

// ---------------- aggregation ----------------

// agg[i][f] = dinv[i]^2 * h[i][f]   (self-loop term, also initializes agg)
__global__ void k_selfloop(const float* __restrict__ h, const float* __restrict__ dinv,
                           float* __restrict__ agg, int logF) {
    int t = blockIdx.x * blockDim.x + threadIdx.x;  // exact N*F threads
    float d = dinv[t >> logF];
    agg[t] = d * d * h[t];
}

// agg[col][f] += norm_e * h[row][f], float4 per thread
__global__ void k_edge_scatter(const float* __restrict__ h, float* __restrict__ agg,
                               const int* __restrict__ row32, const int* __restrict__ col32,
                               const float* __restrict__ nrm, int logC, int F) {
    int t = blockIdx.x * blockDim.x + threadIdx.x;  // exact E * (F/4) threads
    int e = t >> logC;
    int c = (t & ((1 << logC) - 1)) << 2;
    int r = row32[e];
    int cl = col32[e];
    float nm = nrm[e];
    const float4 hv = *reinterpret_cast<const float4*>(h + (size_t)r * F + c);
    float* dst = agg + (size_t)cl * F + c;
    GATOMIC_ADD(dst + 0, nm * hv.x);
    GATOMIC_ADD(dst + 1, nm * hv.y);
    GATOMIC_ADD(dst + 2, nm * hv.z);
    GATOMIC_ADD(dst + 3, nm * hv.w);
}

__global__ void k_bias_relu(float* __restrict__ a, const float* __restrict__ b, int fmask) {
    int t = blockIdx.x * blockDim.x + threadIdx.x;  // exact N*F threads
    float v = a[t] + b[t & fmask];
    a[t] = v > 0.0f ? v : 0.0f;
}

// ---------------- pooling + MLP head ----------------

__global__ void k_zero(float* __restrict__ p, int n) {
    int i = blockIdx.x * blockDim.x + threadIdx.x;
    if (i < n) p[i] = 0.0f;
}

#define POOL_SLOTS (N_GRAPHS * 512 + N_GRAPHS)

__global__ __launch_bounds__(256) void k_pool(const float* __restrict__ act,
                                              const int* __restrict__ batch,
                                              float* __restrict__ gsum) {
    __shared__ float sm[POOL_SLOTS];
    for (int i = threadIdx.x; i < POOL_SLOTS; i += 256) sm[i] = 0.0f;
    __syncthreads();
    int ibeg = blockIdx.x * 64;
    int iend = ibeg + 64;
    if (iend > N_NODES) iend = N_NODES;
    for (int i = ibeg; i < iend; ++i) {
        int g = batch[i];
        const float* ar = act + (size_t)i * 512;
        for (int f = threadIdx.x; f < 512; f += 256)
            atomicAdd(&sm[g * 512 + f], ar[f]);      // ds_add_f32
        if (threadIdx.x == 0) sm[N_GRAPHS * 512 + g] += 1.0f;
    }
    __syncthreads();
    for (int i = threadIdx.x; i < POOL_SLOTS; i += 256) {
        float v = sm[i];
        if (v != 0.0f) GATOMIC_ADD(&gsum[i], v);
    }
}

__global__ __launch_bounds__(128) void k_fc(const float* __restrict__ gsum,
                                            const float* __restrict__ fcw1,
                                            const float* __restrict__ fcb1,
                                            const float* __restrict__ fcw2,
                                            const float* __restrict__ fcb2,
                                            float* __restrict__ out) {
    __shared__ float red[128];
    int g = blockIdx.x;
    int j = threadIdx.x;
    float cnt = gsum[N_GRAPHS * 512 + g];
    float inv = 1.0f / fmaxf(cnt, 1.0f);
    const float* ps = gsum + g * 512;
    float acc = 0.0f;
    for (int k = 0; k < 512; ++k) acc = fmaf(ps[k], fcw1[k * 128 + j], acc);
    float h = acc * inv + fcb1[j];
    h = h > 0.0f ? h : 0.0f;
    red[j] = h * fcw2[j];
    __syncthreads();
    for (int s = 64; s > 0; s >>= 1) {
        if (j < s) red[j] += red[j + s];
        __syncthreads();
    }
    if (j == 0) out[g] = red[0] + fcb2[0];
}

// ---------------- host-side orchestration ----------------

extern "C" void kernel_launch(void* const* d_in, const int* in_sizes, int n_in,
                              void* d_out, int out_size, void* d_ws, size_t ws_size,
                              hipStream_t stream) {
    const float* x    = (const float*)d_in[0];
    const int*   ei   = (const int*)d_in[1];   // [2, E] int32
    const int*   bat  = (const int*)d_in[2];
    const float* W1   = (const float*)d_in[3];
    const float* b1   = (const float*)d_in[4];
    const float* W2   = (const float*)d_in[5];
    const float* b2   = (const float*)d_in[6];
    const float* W3   = (const float*)d_in[7];
    const float* b3   = (const float*)d_in[8];
    const float* fcw1 = (const float*)d_in[9];
    const float* fcb1 = (const float*)d_in[10];
    const float* fcw2 = (const float*)d_in[11];
    const float* fcb2 = (const float*)d_in[12];
    float* out = (float*)d_out;

    const size_t NF = (size_t)N_NODES * 1024;
    float* ws   = (float*)d_ws;
    float* H    = ws;                 // [N,1024] GEMM outputs
    float* A    = ws + NF;            // [N,1024] aggregation / activations
    float* dinv = ws + 2 * NF;        // [N]   (deg, then d^-1/2)
    float* nrm  = dinv + N_NODES;     // [E]
    int*  row32 = (int*)(nrm + N_EDGES);
    int*  col32 = row32 + N_EDGES;
    float* Wt2  = (float*)(col32 + N_EDGES);  // [1024,512]
    float* Wt3  = Wt2 + 512 * 1024;           // [512,1024]
    float* gsum = Wt3 + 1024 * 512;           // [16*512 + 16]

    const int TB = 256;
    const int gN  = (N_NODES + TB - 1) / TB;    // 196
    const int gE  = (N_EDGES + TB - 1) / TB;    // 1563
    const int gNF512  = N_NODES * 512 / TB;     // 100000 (exact)
    const int gNF1024 = N_NODES * 1024 / TB;    // 200000 (exact)
    const int gES512  = N_EDGES * 128 / TB;     // 200000 (exact): E * (512/4)
    const int gES1024 = N_EDGES * 256 / TB;     // 400000 (exact)

    // degrees, norms, edge index conversion
    k_fill1<<<gN, TB, 0, stream>>>(dinv, N_NODES);
    k_edge_prep<<<gE, TB, 0, stream>>>(ei, row32, col32, dinv, N_EDGES);
    k_rsqrt_inplace<<<gN, TB, 0, stream>>>(dinv, N_NODES);
    k_norm<<<gE, TB, 0, stream>>>(dinv, row32, col32, nrm, N_EDGES);

    // weight transposes for WMMA B-operand
    k_transpose<<<512 * 1024 / TB, TB, 0, stream>>>(W2, Wt2, 512, 10);
    k_transpose<<<1024 * 512 / TB, TB, 0, stream>>>(W3, Wt3, 1024, 9);

    // ---- layer 1: 3 -> 512 ----
    k_gemm1<<<gNF512, TB, 0, stream>>>(x, W1, H);
    k_selfloop<<<gNF512, TB, 0, stream>>>(H, dinv, A, 9);
    k_edge_scatter<<<gES512, TB, 0, stream>>>(H, A, row32, col32, nrm, 7, 512);
    k_bias_relu<<<gNF512, TB, 0, stream>>>(A, b1, 511);

    // ---- layer 2: 512 -> 1024 ----
    k_wmma_gemm_f32<<<dim3(N_NODES / 16, 2), TB, 0, stream>>>(A, Wt2, H, 512, 1024);
    k_selfloop<<<gNF1024, TB, 0, stream>>>(H, dinv, A, 10);
    k_edge_scatter<<<gES1024, TB, 0, stream>>>(H, A, row32, col32, nrm, 8, 1024);
    k_bias_relu<<<gNF1024, TB, 0, stream>>>(A, b2, 1023);

    // ---- layer 3: 1024 -> 512 ----
    k_wmma_gemm_f32<<<dim3(N_NODES / 16, 1), TB, 0, stream>>>(A, Wt3, H, 1024, 512);
    k_selfloop<<<gNF512, TB, 0, stream>>>(H, dinv, A, 9);
    k_edge_scatter<<<gES512, TB, 0, stream>>>(H, A, row32, col32, nrm, 7, 512);
    k_bias_relu<<<gNF512, TB, 0, stream>>>(A, b3, 511);

    // ---- global mean pool + MLP head ----
    k_zero<<<(POOL_SLOTS + TB - 1) / TB, TB, 0, stream>>>(gsum, POOL_SLOTS);
    k_pool<<<(N_NODES + 63) / 64, TB, 0, stream>>>(A, bat, gsum);
    k_fc<<<N_GRAPHS, 128, 0, stream>>>(gsum, fcw1, fcb1, fcw2, fcb2, out);
}